// Encoder_56556129354623
// MI455X (gfx1250) — compile-verified
//
#include <hip/hip_runtime.h>
#include <hip/hip_bf16.h>

typedef __attribute__((ext_vector_type(2))) float v2f;
typedef __attribute__((ext_vector_type(8))) float v8f;

// ---------------------------------------------------------------------------
// Build Wcat (HID x 4*DOUT) = [Wm_top | Wm_bot | Ws_top | Ws_bot]
// so that P = hidden @ Wcat gives per-node projections:
//   P[:,   0: 64] = hidden @ Wm_t   (= W_miu[:128])
//   P[:,  64:128] = hidden @ Wm_b   (= W_miu[128:])
//   P[:, 128:192] = hidden @ Ws_t
//   P[:, 192:256] = hidden @ Ws_b
// ---------------------------------------------------------------------------
__global__ void build_wcat_kernel(const float* __restrict__ W_miu,
                                  const float* __restrict__ W_sigma,
                                  float* __restrict__ Wcat,
                                  int HID, int DOUT) {
  int tid = blockIdx.x * blockDim.x + threadIdx.x;
  int PC = 4 * DOUT;
  if (tid >= HID * PC) return;
  int k = tid / PC;
  int j = tid - k * PC;
  float v;
  if (j < DOUT)            v = W_miu[(size_t)k * DOUT + j];
  else if (j < 2 * DOUT)   v = W_miu[(size_t)(HID + k) * DOUT + (j - DOUT)];
  else if (j < 3 * DOUT)   v = W_sigma[(size_t)k * DOUT + (j - 2 * DOUT)];
  else                     v = W_sigma[(size_t)(HID + k) * DOUT + (j - 3 * DOUT)];
  Wcat[tid] = v;
}

// ---------------------------------------------------------------------------
// FP32 WMMA GEMM: C[M,N] = A[M,K] @ B[K,N]   (row-major everywhere)
// One wave32 computes a 16x64 tile with 4 accumulators via
// V_WMMA_F32_16X16X4_F32.  Requires M%16==0, K%4==0, N%64==0.
//
// Fragment layouts (CDNA5 ISA 7.12.2, 32-bit data):
//   A 16x4 : lane L, vgpr i -> A[m=L%16][k0 + 2*(L/16) + i]
//   B 4x16 : lane L, vgpr i -> B[k0 + 2*(L/16) + i][n=cBase + L%16]
//   C 16x16: lane L, vgpr v -> C[m=v + 8*(L/16)][n=L%16]
// ---------------------------------------------------------------------------
__global__ void gemm_f32_wmma_16x64(const float* __restrict__ A,
                                    const float* __restrict__ B,
                                    float* __restrict__ C,
                                    int M, int K, int N) {
  int wave    = blockIdx.x * (blockDim.x >> 5) + (threadIdx.x >> 5);
  int nGroups = N >> 6;
  int mTile   = wave / nGroups;
  int nGroup  = wave - mTile * nGroups;
  if (mTile * 16 >= M) return;               // wave-uniform: EXEC stays all-1s
  int lane = threadIdx.x & 31;
  int r    = lane & 15;
  int h    = lane >> 4;

  const float* Arow = A + (size_t)(mTile * 16 + r) * K + 2 * h;
  const float* Bp0  = B + (size_t)(2 * h) * N + nGroup * 64 + r;

  v8f acc0 = {}, acc1 = {}, acc2 = {}, acc3 = {};
#pragma unroll 4
  for (int k = 0; k < K; k += 4) {
    v2f a = *(const v2f*)(Arow + k);         // b64 load of A fragment
    const float* Bp = Bp0 + (size_t)k * N;
    v2f b0, b1, b2, b3;
    b0.x = Bp[0];  b0.y = Bp[N];
    b1.x = Bp[16]; b1.y = Bp[N + 16];
    b2.x = Bp[32]; b2.y = Bp[N + 32];
    b3.x = Bp[48]; b3.y = Bp[N + 48];
    acc0 = __builtin_amdgcn_wmma_f32_16x16x4_f32(false, a, false, b0, (short)0, acc0, false, false);
    acc1 = __builtin_amdgcn_wmma_f32_16x16x4_f32(false, a, false, b1, (short)0, acc1, false, false);
    acc2 = __builtin_amdgcn_wmma_f32_16x16x4_f32(false, a, false, b2, (short)0, acc2, false, false);
    acc3 = __builtin_amdgcn_wmma_f32_16x16x4_f32(false, a, false, b3, (short)0, acc3, false, false);
  }

  float* Cp = C + (size_t)nGroup * 64 + r;
#pragma unroll
  for (int v = 0; v < 8; ++v) {
    size_t row = (size_t)(mTile * 16 + v + 8 * h) * N;
    Cp[row]      = acc0[v];
    Cp[row + 16] = acc1[v];
    Cp[row + 32] = acc2[v];
    Cp[row + 48] = acc3[v];
  }
}

// ---------------------------------------------------------------------------
// hidden[row] += gcn_values[e] * xw[col]   (segment_sum via f32 atomics)
// ---------------------------------------------------------------------------
__global__ void spmm_scatter_kernel(const float* __restrict__ vals,
                                    const int* __restrict__ idx,   // [2, NNZ]
                                    const float* __restrict__ XW,
                                    float* __restrict__ hidden,
                                    int NNZ, int HID) {
  long long tid = (long long)blockIdx.x * blockDim.x + threadIdx.x;
  if (tid >= (long long)NNZ * HID) return;
  int e = (int)(tid / HID);
  int j = (int)(tid - (long long)e * HID);
  int row = idx[e];         // segment id  (gcn_indices[0])
  int col = idx[NNZ + e];   // source node (gcn_indices[1])
  float m = vals[e] * XW[(size_t)col * HID + j];
  __hip_atomic_fetch_add(&hidden[(size_t)row * HID + j], m,
                         __ATOMIC_RELAXED, __HIP_MEMORY_SCOPE_AGENT);
}

// ---------------------------------------------------------------------------
// Directed edges: gather projections, reparameterize, write edge_in/edge_out
// ---------------------------------------------------------------------------
__global__ void edge_kernel(const float* __restrict__ P,
                            const int* __restrict__ edges,  // [2, E]
                            const float* __restrict__ noise_in,
                            const float* __restrict__ noise_out,
                            float* __restrict__ out_in,
                            float* __restrict__ out_out,
                            int E, int DOUT) {
  long long tid = (long long)blockIdx.x * blockDim.x + threadIdx.x;
  if (tid >= (long long)E * DOUT) return;
  int e = (int)(tid / DOUT);
  int j = (int)(tid - (long long)e * DOUT);
  int s = edges[e];
  int d = edges[E + e];
  const float* Ps = P + (size_t)s * (4 * DOUT);
  const float* Pd = P + (size_t)d * (4 * DOUT);
  float miu_in  = Ps[j]            + Pd[DOUT + j];
  float ls_in   = Ps[2 * DOUT + j] + Pd[3 * DOUT + j];
  float miu_out = Pd[j]            + Ps[DOUT + j];
  float ls_out  = Pd[2 * DOUT + j] + Ps[3 * DOUT + j];
  size_t o = (size_t)e * DOUT + j;
  out_in[o]  = noise_in[o]  * expf(ls_in)  + miu_in;
  out_out[o] = noise_out[o] * expf(ls_out) + miu_out;
}

// ---------------------------------------------------------------------------
// Bi edges: accumulate node_miu and node_var with f32 atomics
// ---------------------------------------------------------------------------
__global__ void bi_kernel(const float* __restrict__ P,
                          const int* __restrict__ edges,  // [2, BIE]
                          const float* __restrict__ bvals,
                          float* __restrict__ node_miu,
                          float* __restrict__ node_var,
                          int BIE, int DOUT) {
  long long tid = (long long)blockIdx.x * blockDim.x + threadIdx.x;
  if (tid >= (long long)BIE * DOUT) return;
  int e = (int)(tid / DOUT);
  int j = (int)(tid - (long long)e * DOUT);
  int b0 = edges[e];
  int b1 = edges[BIE + e];
  const float* P0 = P + (size_t)b0 * (4 * DOUT);
  const float* P1 = P + (size_t)b1 * (4 * DOUT);
  float v   = bvals[e];
  float miu = P0[j]            + P1[DOUT + j];
  float ls  = P0[2 * DOUT + j] + P1[3 * DOUT + j];
  size_t o = (size_t)b0 * DOUT + j;
  __hip_atomic_fetch_add(&node_miu[o], v * miu,
                         __ATOMIC_RELAXED, __HIP_MEMORY_SCOPE_AGENT);
  __hip_atomic_fetch_add(&node_var[o], v * v * expf(ls),
                         __ATOMIC_RELAXED, __HIP_MEMORY_SCOPE_AGENT);
}

// ---------------------------------------------------------------------------
// node_embed = noise_node * exp(0.5*log(var)) + node_miu  ==  noise*sqrt(var)+miu
// ---------------------------------------------------------------------------
__global__ void node_kernel(const float* __restrict__ node_miu,
                            const float* __restrict__ node_var,
                            const float* __restrict__ noise_node,
                            float* __restrict__ out, long long total) {
  long long tid = (long long)blockIdx.x * blockDim.x + threadIdx.x;
  if (tid >= total) return;
  out[tid] = noise_node[tid] * sqrtf(node_var[tid]) + node_miu[tid];
}

extern "C" void kernel_launch(void* const* d_in, const int* in_sizes, int n_in,
                              void* d_out, int out_size, void* d_ws, size_t ws_size,
                              hipStream_t stream) {
  const float* X          = (const float*)d_in[0];
  const float* W_gcn      = (const float*)d_in[1];
  const float* W_miu      = (const float*)d_in[2];
  const float* W_sigma    = (const float*)d_in[3];
  const float* gcn_values = (const float*)d_in[4];
  const float* bi_values  = (const float*)d_in[5];
  const float* noise_in   = (const float*)d_in[6];
  const float* noise_out  = (const float*)d_in[7];
  const float* noise_node = (const float*)d_in[8];
  const int*   gcn_idx    = (const int*)d_in[9];
  const int*   di_edges   = (const int*)d_in[10];
  const int*   bi_edges   = (const int*)d_in[11];

  const int DOUT = 64;
  const int N    = in_sizes[8] / DOUT;   // noise_node is N x DOUT
  const int DIN  = in_sizes[0] / N;      // 256
  const int HID  = in_sizes[1] / DIN;    // 128
  const int NNZ  = in_sizes[4];
  const int E    = in_sizes[6] / DOUT;
  const int BIE  = in_sizes[5];
  const int PC   = 4 * DOUT;             // 256 projection columns

  // Workspace layout (floats)
  float* ws = (float*)d_ws;
  size_t o = 0;
  float* xw     = ws + o; o += (size_t)N * HID;
  float* hidden = ws + o; o += (size_t)N * HID;
  float* Wcat   = ws + o; o += (size_t)HID * PC;
  float* P      = ws + o; o += (size_t)N * PC;
  float* nmiu   = ws + o; o += (size_t)N * DOUT;
  float* nvar   = ws + o; o += (size_t)N * DOUT;

  hipMemsetAsync(hidden, 0, sizeof(float) * (size_t)N * HID, stream);
  hipMemsetAsync(nmiu,   0, sizeof(float) * (size_t)N * DOUT, stream);
  hipMemsetAsync(nvar,   0, sizeof(float) * (size_t)N * DOUT, stream);

  // 0) concatenated projection weights
  build_wcat_kernel<<<(HID * PC + 255) / 256, 256, 0, stream>>>(
      W_miu, W_sigma, Wcat, HID, DOUT);

  // 1) xw = X @ W_gcn   (WMMA f32)
  {
    int waves = (N / 16) * (HID / 64);
    gemm_f32_wmma_16x64<<<(waves + 7) / 8, 256, 0, stream>>>(
        X, W_gcn, xw, N, DIN, HID);
  }

  // 2) hidden = segment_sum(gcn_values * xw[gather])
  {
    long long t = (long long)NNZ * HID;
    spmm_scatter_kernel<<<(int)((t + 255) / 256), 256, 0, stream>>>(
        gcn_values, gcn_idx, xw, hidden, NNZ, HID);
  }

  // 3) P = hidden @ Wcat   (WMMA f32) -> per-node [mt|mb|st|sb] projections
  {
    int waves = (N / 16) * (PC / 64);
    gemm_f32_wmma_16x64<<<(waves + 7) / 8, 256, 0, stream>>>(
        hidden, Wcat, P, N, HID, PC);
  }

  float* out_node = (float*)d_out;
  float* out_in   = out_node + (size_t)N * DOUT;
  float* out_out  = out_in + (size_t)E * DOUT;

  // 4) directed edges
  {
    long long t = (long long)E * DOUT;
    edge_kernel<<<(int)((t + 255) / 256), 256, 0, stream>>>(
        P, di_edges, noise_in, noise_out, out_in, out_out, E, DOUT);
  }

  // 5) bi edges -> node accumulators
  {
    long long t = (long long)BIE * DOUT;
    bi_kernel<<<(int)((t + 255) / 256), 256, 0, stream>>>(
        P, bi_edges, bi_values, nmiu, nvar, BIE, DOUT);
  }

  // 6) node reparameterization
  {
    long long t = (long long)N * DOUT;
    node_kernel<<<(int)((t + 255) / 256), 256, 0, stream>>>(
        nmiu, nvar, noise_node, out_node, t);
  }
}